// Kernel_11950610535260447599_53472342835848
// MI455X (gfx1250) — compile-verified
//
#include <hip/hip_runtime.h>

typedef float v2f __attribute__((ext_vector_type(2)));
typedef float v8f __attribute__((ext_vector_type(8)));

#define WMMA_F32X4(A, B, Cacc) \
    __builtin_amdgcn_wmma_f32_16x16x4_f32(false, (A), false, (B), (short)0, (Cacc), false, false)

#define S_WAIT_ASYNC0() asm volatile("s_wait_asynccnt 0x0" ::: "memory")

// ---------------------------------------------------------------------------
// Repack w4 (co, ci, kh, kw) -> w4p[(kh*7+kw)][ci>>1][co][ci&1], negated.
// Makes the WMMA A-fragment a single coalesced 8-byte load per lane, and
// folds conv(-x, w) = conv(x, -w) so the LDS input stage is a pure copy.
// ---------------------------------------------------------------------------
__global__ __launch_bounds__(256, 1)
void repack_w4_kernel(const float* __restrict__ w4, float* __restrict__ w4p)
{
    const int o = blockIdx.x * 256 + threadIdx.x;
    if (o >= 49 * 32 * 64 * 2) return;
    const int par = o & 1;
    const int co  = (o >> 1) & 63;
    const int cp  = (o >> 7) & 31;
    const int khw = o >> 12;
    const int ci  = cp * 2 + par;
    w4p[o] = -w4[co * 3136 + ci * 49 + khw];
}

// ---------------------------------------------------------------------------
// Stage A: t4 = conv2d(-x, w4), 7x7, pad 3, 64->64 ch, fp32 WMMA implicit GEMM.
// One workgroup per (n, h): M=co(64), N=w(64), K=(ci,kh,kw)=3136.
// Input staged in LDS in 8-channel chunks via double-buffered
// GLOBAL_LOAD_ASYNC_TO_LDS_B32 (ASYNCcnt), pair-interleaved
// [ci>>1][kh][wp][ci&1] so each B fragment is one ds_load_b64.
// Pad regions pre-zeroed once; asyncs only write the valid interior.
// ---------------------------------------------------------------------------
__global__ __launch_bounds__(256, 1)
void conv7x7_wmma_kernel(const float* __restrict__ x,
                         const float* __restrict__ w4p,
                         float* __restrict__ t4)
{
    // two ping-pong buffers: [buf(2)][cl2(4)][kh(7)][wp(72)][par(2)] floats
    __shared__ float lds[2 * 4 * 7 * 72 * 2];   // 2 x 16128 B

    const int bid  = blockIdx.x;
    const int nb   = bid >> 6;          // batch index
    const int h    = bid & 63;          // output row
    const int tid  = threadIdx.x;
    const int wave = tid >> 5;          // 8 wave32 waves
    const int lane = tid & 31;
    const int ml   = lane & 15;         // M/N index within fragment
    const int half = lane >> 4;         // 0: K=0,1  1: K=2,3

    const int co0 = (wave & 3) << 4;    // co tile base
    const int n0  = (wave >> 2) << 5;   // first w tile base (2 tiles per wave)

    const float* xb = x + nb * 64 * 64 * 64;    // uniform per block -> SGPR pair

    // one-time zero of both buffers (pad columns / OOB-h rows stay zero)
    for (int j = tid; j < 2 * 4 * 7 * 72; j += 256)
        *(float2*)&lds[j * 2] = make_float2(0.0f, 0.0f);
    __syncthreads();

    // issue async fill of chunk cc (8 channels) into buffer `buf`
    auto issue_fill = [&](int cc, int buf) {
        const unsigned bufbase = (unsigned)buf * (4 * 7 * 72 * 2 * 4);  // bytes
        for (int j = tid; j < 4 * 7 * 64; j += 256) {
            const int cl2 = j / 448;            // 7*64
            const int r   = j - cl2 * 448;
            const int kh  = r / 64;
            const int w   = r - kh * 64;
            const int hr  = h + kh - 3;
            if ((unsigned)hr < 64u) {
                // even channel of the pair; odd channel is +16384 B in global
                const unsigned voff =
                    (unsigned)((((cc * 8 + cl2 * 2) * 64 + hr) * 64 + w) * 4);
                const unsigned la =
                    bufbase + (unsigned)((((cl2 * 7 + kh) * 72 + (w + 3)) * 2) * 4);
                asm volatile("global_load_async_to_lds_b32 %0, %1, %2"
                             :: "v"(la), "v"(voff), "s"(xb) : "memory");
                asm volatile("global_load_async_to_lds_b32 %0, %1, %2 offset:16384"
                             :: "v"(la + 4u), "v"(voff), "s"(xb) : "memory");
            }
        }
    };

    v8f acc0 = {};
    v8f acc1 = {};

    issue_fill(0, 0);
    for (int cc = 0; cc < 8; ++cc) {            // 8-channel chunks of ci
        S_WAIT_ASYNC0();                        // own wave's async fills done
        __syncthreads();                        // all waves' fills done
        if (cc < 7) issue_fill(cc + 1, (cc + 1) & 1);
        const float* L = lds + (cc & 1) * (4 * 7 * 72 * 2);

        for (int kh = 0; kh < 7; ++kh) {
            for (int kw = 0; kw < 7; ++kw) {
                // A: w4p[((khw*32 + cc*4 + cl2)*64 + co)*2 + par]
                const float* ap =
                    w4p + (((kh * 7 + kw) * 32 + cc * 4) * 64 + (co0 + ml)) * 2;
                const int bcol = n0 + ml + kw;
#pragma unroll
                for (int kc = 0; kc < 2; ++kc) {
                    const int cl2 = kc * 2 + half;
                    const v2f a = *(const v2f*)(ap + cl2 * 128);
                    const float* bp = &L[((cl2 * 7 + kh) * 72 + bcol) * 2];
                    const v2f b0 = *(const v2f*)bp;
                    const v2f b1 = *(const v2f*)(bp + 32);
                    acc0 = WMMA_F32X4(a, b0, acc0);     // tile at n0
                    acc1 = WMMA_F32X4(a, b1, acc1);     // tile at n0+16 (A reused)
                }
            }
        }
    }

    // C/D layout: VGPR j, lanes 0-15 -> M=j, lanes 16-31 -> M=j+8; N = ml
    const int wr0 = n0 + ml;
#pragma unroll
    for (int j = 0; j < 8; ++j) {
        const int co = co0 + j + (half << 3);
        float* row = t4 + (((nb * 64) + co) * 64 + h) * 64;
        row[wr0]      = acc0[j];
        row[wr0 + 16] = acc1[j];
    }
}

// ---------------------------------------------------------------------------
// Stage B: t8 = 1x1 conv of rolled x (WMMA GEMM, M=co, K=ci=64, N=w) fused
// with depthwise (1,3) dilation-2 conv on t4, t6 = x*t5, out = t6 + t8.
// LDS pair-interleaved [ci>>1][w][ci&1] -> B fragment = 1x ds_load_b64.
// ---------------------------------------------------------------------------
__global__ __launch_bounds__(256, 1)
void fused_dw_1x1_wmma_kernel(const float* __restrict__ x,
                              const float* __restrict__ w5,
                              const float* __restrict__ w8,
                              const float* __restrict__ t4,
                              float* __restrict__ out)
{
    __shared__ float lds[32 * 64 * 2];  // x[n, :, (h-1)%64, :], pair-interleaved

    const int bid = blockIdx.x;
    const int nb  = bid >> 6;
    const int h   = bid & 63;
    const int hm1 = (h + 63) & 63;      // net roll of +1 along H
    const int tid = threadIdx.x;

    for (int j = tid; j < 2048; j += 256) {
        const int ci2 = j >> 6, w = j & 63;
        const float* px = x + (((nb * 64) + ci2 * 2) * 64 + hm1) * 64 + w;
        float2 v;
        v.x = px[0];
        v.y = px[4096];
        *(float2*)&lds[j * 2] = v;
    }
    __syncthreads();

    const int wave = tid >> 5;
    const int lane = tid & 31;
    const int ml   = lane & 15;
    const int half = lane >> 4;
    const int koff = half << 1;
    const int co0  = (wave & 3) << 4;
    const int n0   = (wave >> 2) << 5;

    v8f acc0 = {};
    v8f acc1 = {};

#pragma unroll 4
    for (int k = 0; k < 64; k += 4) {
        const int ci2 = (k >> 1) + half;
        const v2f a = *(const v2f*)(w8 + (co0 + ml) * 64 + k + koff);
        const float* bp = &lds[(ci2 * 64 + n0 + ml) * 2];
        const v2f b0 = *(const v2f*)bp;
        const v2f b1 = *(const v2f*)(bp + 32);
        acc0 = WMMA_F32X4(a, b0, acc0);
        acc1 = WMMA_F32X4(a, b1, acc1);
    }

    // fused epilogue: t5 depthwise on t4, t6 = x * t5, out = t6 + t8
#pragma unroll
    for (int j = 0; j < 8; ++j) {
        const int co = co0 + j + (half << 3);
        const float* t4row = t4 + (((nb * 64) + co) * 64 + h) * 64;
        const float* xrow  = x  + (((nb * 64) + co) * 64 + h) * 64;
        float* orow        = out + (((nb * 64) + co) * 64 + h) * 64;
        const float w50 = w5[co * 3 + 0];
        const float w51 = w5[co * 3 + 1];
        const float w52 = w5[co * 3 + 2];
        {
            const int wp = n0 + ml;
            const float tm2 = (wp >= 2)  ? t4row[wp - 2] : 0.0f;
            const float tc  = t4row[wp];
            const float tp2 = (wp <= 61) ? t4row[wp + 2] : 0.0f;
            const float t5v = w50 * tm2 + w51 * tc + w52 * tp2;
            orow[wp] = xrow[wp] * t5v + acc0[j];
        }
        {
            const int wp = n0 + ml + 16;
            const float tm2 = t4row[wp - 2];                // wp >= 16: in range
            const float tc  = t4row[wp];
            const float tp2 = (wp <= 61) ? t4row[wp + 2] : 0.0f;
            const float t5v = w50 * tm2 + w51 * tc + w52 * tp2;
            orow[wp] = xrow[wp] * t5v + acc1[j];
        }
    }
}

extern "C" void kernel_launch(void* const* d_in, const int* in_sizes, int n_in,
                              void* d_out, int out_size, void* d_ws, size_t ws_size,
                              hipStream_t stream) {
    const float* x  = (const float*)d_in[0];   // (16,64,64,64)
    const float* w4 = (const float*)d_in[1];   // (64,64,7,7)
    const float* w5 = (const float*)d_in[2];   // (64,1,1,3)
    const float* w8 = (const float*)d_in[3];   // (64,64,1,1)
    float* out = (float*)d_out;                // (16,64,64,64)

    float* t4  = (float*)d_ws;                       // 16 MiB
    float* w4p = (float*)d_ws + 16 * 64 * 64 * 64;   // + 784 KiB repacked weights

    dim3 block(256);                // 8 wave32 waves
    repack_w4_kernel<<<dim3((49 * 32 * 64 * 2 + 255) / 256), block, 0, stream>>>(w4, w4p);
    conv7x7_wmma_kernel<<<dim3(16 * 64), block, 0, stream>>>(x, w4p, t4);
    fused_dw_1x1_wmma_kernel<<<dim3(16 * 64), block, 0, stream>>>(x, w5, w8, t4, out);
}